// ImageWiseConv2d_60309930770454
// MI455X (gfx1250) — compile-verified
//
#include <hip/hip_runtime.h>

// MI455X / gfx1250, wave32. Per-sample grouped conv, fp32, via V_WMMA_F32_16X16X4_F32.
//
// Tap-parallel implicit GEMM:
//   D[m=pixel, t=tap] = sum_c img[c, pixel_m] * w[c, tap_t]      (K = channels)
//   out[y,x]          = sum_t D[(y,x)+tap_t, t]                  (9-way LDS gather)
//
// Memory-bound workload (272 MB @ 23.3 TB/s ~= 12 us floor); this reads images
// ~1.125x (row halo), keeps MACs on the f32 WMMA pipe (~1.16M wmma ops total).

typedef __attribute__((ext_vector_type(2))) float v2f;
typedef __attribute__((ext_vector_type(8))) float v8f;

constexpr int Wd = 128, Hd = 128, Cd = 64;
constexpr int OW = 126, OH = 126;
constexpr int RMAX = 18;      // input rows staged per block (16 output rows + 2 halo)
constexpr int TS = 9;         // taps stored per pixel in LDS (stride 9 words -> conflict-free)

__global__ __launch_bounds__(256)
void iwconv_wmma_f32(const float* __restrict__ images,
                     const float* __restrict__ kernels,
                     float* __restrict__ out)
{
    extern __shared__ float Dl[];              // [R][128][9] tap partials
    const int n    = blockIdx.y;
    const int y0   = blockIdx.x * 16;
    const int R    = min(RMAX, Hd - y0);       // input rows this block
    const int Ro   = min(16,   OH - y0);       // output rows this block
    const int tidx = threadIdx.x;
    const int lane = tidx & 31;
    const int wave = tidx >> 5;
    const int hi   = lane >> 4;                // half-wave: selects K pair {0,1} vs {2,3}
    const int li   = lane & 15;

    const float* img = images  + (size_t)n * Cd * Hd * Wd;
    const float* krn = kernels + (size_t)n * Cd * 9;

    // ---- B matrix (weights) in registers, shared by every tile of this sample ----
    // B is 4x16 (KxN): v0 = rows K=0|K=2 (lanes 0-15 | 16-31), v1 = rows K=1|K=3.
    // Column t (N) = filter tap kh*3+kw; taps 9..15 are zero padding.
    // Single predicate around all 32 loads -> one exec save/restore, one clause.
    v2f bw[16];
#pragma unroll
    for (int k = 0; k < 16; ++k) { bw[k].x = 0.f; bw[k].y = 0.f; }
    if (li < TS) {
#pragma unroll
        for (int k = 0; k < 16; ++k) {
            const int c0 = 4 * k + 2 * hi;
            bw[k].x = krn[(c0    ) * 9 + li];
            bw[k].y = krn[(c0 + 1) * 9 + li];
        }
    }

    // ---- phase 1: per-wave 16-column strip, loop over rows; K-chain of 16 WMMAs ----
    const int xt = wave * 16;                  // strip origin (8 waves * 16 = 128 cols)
    const int xl = xt + li;                    // this lane's pixel column (M index)
    for (int r = 0; r < R; ++r) {
        const int y = y0 + r;
        // A is 16x4 (MxK): v0 = K=0|K=2, v1 = K=1|K=3; M = lane&15 -> pixel x.
        const float* p = img + (size_t)(2 * hi) * (Hd * Wd) + (size_t)y * Wd + xl;
        v2f a[16];
#pragma unroll
        for (int k = 0; k < 16; ++k) {
            a[k].x = p[(size_t)(4 * k    ) * (Hd * Wd)];
            a[k].y = p[(size_t)(4 * k + 1) * (Hd * Wd)];
        }
        if (r + 1 < R) __builtin_prefetch(p + Wd, 0, 3);   // next row, same channels

        v8f acc = {};
#pragma unroll
        for (int k = 0; k < 16; ++k) {
            // (neg_a, A, neg_b, B, c_mod, C, reuse_a, reuse_b)
            acc = __builtin_amdgcn_wmma_f32_16x16x4_f32(
                false, a[k], false, bw[k], (short)0, acc, false, false);
        }

        // D layout: lane holds column t=li for rows m = hi*8 + j (j = vgpr index).
        if (li < TS) {
            float* q = &Dl[((size_t)r * Wd + xt + hi * 8) * TS + li];
#pragma unroll
            for (int j = 0; j < 8; ++j) q[j * TS] = acc[j];
        }
    }
    __syncthreads();

    // ---- phase 2: gather 9 taps per output pixel (no integer division) ----
    // xo = tidx & 127 (stores coalesced over 126 cols), two rows in flight.
    float* outp = out + (size_t)n * OH * OW;
    const int xo = tidx & 127;
    if (xo < OW) {
        for (int yo = tidx >> 7; yo < Ro; yo += 2) {
            float s = 0.f;
#pragma unroll
            for (int kh = 0; kh < 3; ++kh)
#pragma unroll
                for (int kw = 0; kw < 3; ++kw)
                    s += Dl[((size_t)(yo + kh) * Wd + xo + kw) * TS + (kh * 3 + kw)];
            outp[(size_t)(y0 + yo) * OW + xo] = s;
        }
    }
}

extern "C" void kernel_launch(void* const* d_in, const int* in_sizes, int n_in,
                              void* d_out, int out_size, void* d_ws, size_t ws_size,
                              hipStream_t stream) {
    const float* images  = (const float*)d_in[0];   // [64,64,128,128] f32
    const float* kernels = (const float*)d_in[1];   // [64,64,3,3]     f32
    float*       out     = (float*)d_out;           // [64,1,126,126]  f32

    dim3 grid((OH + 15) / 16, 64);                  // 8 row-blocks x 64 samples
    dim3 block(256);                                // 8 wave32s
    size_t lds = (size_t)RMAX * Wd * TS * sizeof(float);  // 82944 B < 320 KB
    hipLaunchKernelGGL(iwconv_wmma_f32, grid, block, lds, stream,
                       images, kernels, out);
}